// MPSWavefunction_3556232921368
// MI455X (gfx1250) — compile-verified
//
#include <hip/hip_runtime.h>

// MPS wavefunction contraction for MI455X (gfx1250), wave32 + WMMA + TDM.
//
// Per wave: a 16-row batch tile. v (16x128 f32) lives in LDS (row-padded 132).
// Per site: the Tensor Data Mover DMAs the full [A0;A1] pair (256x128 f32,
// 128KB) into LDS (double-buffered, padded to 136 floats/row via the TDM pad
// feature), overlapping the previous site's WMMA sweep. All 4 waves share the
// staged matrices. v_new = sel0(v)@A0 + sel1(v)@A1 via chained
// V_WMMA_F32_16X16X4_F32 (native CDNA5 f32 matrix op -> full f32 numerics).

typedef float v2f __attribute__((ext_vector_type(2)));
typedef float v8f __attribute__((ext_vector_type(8)));
typedef unsigned int v4u __attribute__((ext_vector_type(4)));
typedef int v4i __attribute__((ext_vector_type(4)));
typedef int v8i __attribute__((ext_vector_type(8)));

#define NSITES 64
#define BOND   128
#define BATCH  16384
#define VROWPAD 132             // v panel row stride (floats): conflict-free
#define AROWPAD 136             // staged A row stride: 128 + 8 dword TDM pad
#define WAVES_PER_BLOCK 4
#define ROWS_PER_WAVE   16
#define THREADS (WAVES_PER_BLOCK * 32)
#define BLOCKS  (BATCH / (WAVES_PER_BLOCK * ROWS_PER_WAVE))   // 256
#define ABUF_FLOATS (2 * BOND * AROWPAD)      // one site: A0 rows + A1 rows
#define LDS_FLOATS  (2 * ABUF_FLOATS + WAVES_PER_BLOCK * ROWS_PER_WAVE * VROWPAD)
#define LDS_BYTES   (LDS_FLOATS * 4)          // 312320 B <= 320KB WGP LDS

extern __shared__ float ldsv[];

// Issue one TDM load: [A0;A1] of site s (256 rows x 128 f32, row stride 128)
// -> LDS at byte offset lds_off, with 8-dword padding every 128 dwords
// (=> 136-float LDS row stride). D# per CDNA5 ISA ch.8.
__device__ __forceinline__ void tdm_load_site(const float* __restrict__ data,
                                              int s, unsigned lds_off) {
    unsigned long long g =
        (unsigned long long)(uintptr_t)(data + (size_t)(2 * s) * BOND * BOND);
    v4u g0;
    g0.x = 1u;                                   // count=1, user descriptor
    g0.y = lds_off;                              // lds_addr (bytes)
    g0.z = (unsigned)(g & 0xFFFFFFFFull);        // global_addr[31:0]
    g0.w = (unsigned)((g >> 32) & 0x01FFFFFFull) // global_addr[56:32]
         | 0x80000000u;                          // type=2 ("image")
    v8i g1;
    g1[0] = (2 << 16)      // data_size = 4 bytes
          | (1 << 20)      // pad_enable
          | (6 << 22)      // pad_interval code 6 = every 128 dwords
          | (7 << 25);     // pad_amount  code 7 = 8 dwords
    g1[1] = (BOND & 0xFFFF) << 16;          // tensor_dim0 = 128
    g1[2] = ((2 * BOND) & 0xFFFF) << 16;    // dim0 hi=0 | tensor_dim1 = 256
    g1[3] = (BOND & 0xFFFF) << 16;          // dim1 hi=0 | tile_dim0 = 128
    g1[4] = (2 * BOND);                     // tile_dim1 = 256 | tile_dim2 = 0
    g1[5] = BOND;                           // tensor_dim0_stride = 128
    g1[6] = 0;                              // stride hi | tensor_dim1_stride lo
    g1[7] = 0;
    v4i zz4 = {0, 0, 0, 0};                 // groups 2/3 unused (2-D tensor)
    v8i zz8 = {0, 0, 0, 0, 0, 0, 0, 0};     // 6-arg form's extra group
    __builtin_amdgcn_tensor_load_to_lds(g0, g1, zz4, zz4, zz8, 0);
}

__global__ __launch_bounds__(THREADS, 1)
void mps_wmma_kernel(const int*   __restrict__ onstate,   // [BATCH][NSITES]
                     const float* __restrict__ data,      // [NSITES][2][BOND][BOND]
                     const float* __restrict__ left,      // [BOND]
                     const float* __restrict__ right,     // [BOND]
                     const int*   __restrict__ image2,    // [NSITES]
                     float*       __restrict__ out)       // [BATCH]
{
    const int lane  = threadIdx.x & 31;
    const int wave  = threadIdx.x >> 5;
    const int row16 = lane & 15;    // batch row within tile
    const int half  = lane >> 4;    // lane group 0/1
    float* astage = ldsv;                                     // 2 site buffers
    float* vbuf   = ldsv + 2 * ABUF_FLOATS + wave * (ROWS_PER_WAVE * VROWPAD);
    const int batch0 = (blockIdx.x * WAVES_PER_BLOCK + wave) * ROWS_PER_WAVE;
    const unsigned astage_base = (unsigned)(uintptr_t)astage; // LDS byte offset

    // Kick off the DMA for site 0 into buffer 0 before anything else.
    if (wave == 0) tdm_load_site(data, 0, astage_base);

    // v0 = broadcast(left) into this wave's LDS panel
    for (int idx = lane; idx < ROWS_PER_WAVE * BOND; idx += 32) {
        const int r = idx >> 7;
        const int c = idx & (BOND - 1);
        vbuf[r * VROWPAD + c] = left[c];
    }

    for (int s = 0; s < NSITES; ++s) {
        if (wave == 0) __builtin_amdgcn_s_wait_tensorcnt(0);  // buf[s&1] landed
        __syncthreads();   // data visible to all waves; prev compute finished
        if (wave == 0 && s + 1 < NSITES)                      // prefetch s+1
            tdm_load_site(data, s + 1,
                          astage_base + (unsigned)(((s + 1) & 1) * ABUF_FLOATS * 4));

        const float* As0 = astage + (s & 1) * ABUF_FLOATS;    // A0 rows 0..127
        const float* As1 = As0 + BOND * AROWPAD;              // A1 rows 0..127
        const int occ = onstate[(batch0 + row16) * NSITES + image2[s]]; // 0/1

        v8f acc[8];
        #pragma unroll
        for (int j = 0; j < 8; ++j)
            acc[j] = (v8f)(0.0f);

        // K sweep in chunks of 4 (WMMA f32 16x16x4): D += a0*B0 + a1*B1
        for (int k = 0; k < BOND / 4; ++k) {
            const int kbase = k * 4 + half * 2;  // A-frag lane K = kbase,kbase+1
            const v2f vf = *(const v2f*)&vbuf[row16 * VROWPAD + kbase];

            v2f a0, a1;
            a0.x = occ ? 0.0f : vf.x;  a0.y = occ ? 0.0f : vf.y;
            a1.x = occ ? vf.x : 0.0f;  a1.y = occ ? vf.y : 0.0f;

            // B-frag: VGPR g, lane -> row kbase + g, col j*16 + row16.
            // Paired rows are 136 dwords apart -> ds_load_2addr, conflict-free.
            const float* pb0 = As0 + (size_t)kbase * AROWPAD + row16;
            const float* pb1 = As1 + (size_t)kbase * AROWPAD + row16;

            #pragma unroll
            for (int j = 0; j < 8; ++j) {
                v2f b0, b1;
                b0.x = pb0[j * 16];
                b0.y = pb0[j * 16 + AROWPAD];
                b1.x = pb1[j * 16];
                b1.y = pb1[j * 16 + AROWPAD];
                acc[j] = __builtin_amdgcn_wmma_f32_16x16x4_f32(
                    false, a0, false, b0, (short)0, acc[j], false, false);
                acc[j] = __builtin_amdgcn_wmma_f32_16x16x4_f32(
                    false, a1, false, b1, (short)0, acc[j], false, false);
            }
        }

        // v_new back to LDS: C/D layout lane -> row g + 8*half, col j*16+row16
        #pragma unroll
        for (int j = 0; j < 8; ++j) {
            #pragma unroll
            for (int g = 0; g < 8; ++g) {
                const int r = g + 8 * half;
                vbuf[r * VROWPAD + j * 16 + row16] = acc[j][g];
            }
        }
        // buf[(s+2)&1] overwrite is guarded by the barrier at loop top
    }

    // out[b] = v[row] . right  (2 lanes per row, 64 cols each, shfl combine)
    float sum = 0.0f;
    const int cbase = half * 64;
    for (int c = 0; c < 64; ++c) {
        const int col = cbase + c;
        sum += vbuf[row16 * VROWPAD + col] * right[col];
    }
    sum += __shfl_xor(sum, 16, 32);
    if (half == 0)
        out[batch0 + row16] = sum;
}

extern "C" void kernel_launch(void* const* d_in, const int* in_sizes, int n_in,
                              void* d_out, int out_size, void* d_ws, size_t ws_size,
                              hipStream_t stream) {
    const int*   onstate = (const int*)  d_in[0];
    const float* data    = (const float*)d_in[1];
    const float* left    = (const float*)d_in[2];
    const float* right   = (const float*)d_in[3];
    const int*   image2  = (const int*)  d_in[4];
    float*       out     = (float*)d_out;

    // Raise dynamic-LDS cap (gfx1250 WGP has 320KB). Host-side, idempotent,
    // graph-capture safe (no alloc/sync involved).
    (void)hipFuncSetAttribute((const void*)mps_wmma_kernel,
                              hipFuncAttributeMaxDynamicSharedMemorySize,
                              LDS_BYTES);
    mps_wmma_kernel<<<dim3(BLOCKS), dim3(THREADS), LDS_BYTES, stream>>>(
        onstate, data, left, right, image2, out);
}